// ComplexFusionModule_17222818857086
// MI455X (gfx1250) — compile-verified
//
#include <hip/hip_runtime.h>
#include <hip/hip_bf16.h>

// dims (compile-time constants from the reference)
#define BB   32
#define C1   4
#define C2   64
#define HH   16
#define WW   16
#define KK   3
#define INF_ 1024          // C1*H*W
#define OUTF 36864         // C2*C2*K*K
#define PIX  256           // H*W

typedef float v2f __attribute__((ext_vector_type(2)));
typedef float v4f __attribute__((ext_vector_type(4)));
typedef float v8f __attribute__((ext_vector_type(8)));

// ---------------------------------------------------------------------------
// Kernel 1: k_sum[b,o] = sum_{g,i} dw[b,g,i]*kg_w[g,o,i] + sum_g kg_b[g,o]
// GEMM  M=32 (b), N=36864 (o), K=4096 (g*1024+i), NT layout (both K-contig).
// One block = 64-wide N strip, all M. 8 waves -> 8 C-tiles (2 Mt x 4 Nt).
// ---------------------------------------------------------------------------
#define LDA 68   // padded LDS row stride in dwords (odd*4 -> 16B-aligned rows, skewed banks)

__global__ __launch_bounds__(256) void ksum_gemm_wmma(
    const float* __restrict__ x1, const float* __restrict__ kgw,
    const float* __restrict__ kgb, float* __restrict__ ksum)
{
  __shared__ float Abuf[32 * LDA];   // 32 x 64 A chunk
  __shared__ float Bbuf[64 * LDA];   // 64 x 64 B^T chunk (rows = n, cols = k)

  const int t    = threadIdx.x;
  const int n0   = blockIdx.x * 64;
  const int wave = t >> 5;           // wave32
  const int lane = t & 31;
  const int half = lane >> 4;        // lanes 16..31 carry K+2,K+3 of each wmma step
  const int l15  = lane & 15;
  const int mt   = wave & 1;         // M tile (0/1)
  const int nt   = wave >> 1;        // N tile (0..3)

  const int arow = mt * 16 + l15;    // A fragment row (M)
  const int brow = nt * 16 + l15;    // B fragment row (N)

  v8f acc = {0.f, 0.f, 0.f, 0.f, 0.f, 0.f, 0.f, 0.f};

  for (int kc = 0; kc < 64; ++kc) {          // K chunks of 64 (4096 total)
    const int g  = kc >> 4;                  // kg_w group (1024 k per group)
    const int i0 = (kc & 15) * 64;           // offset within group

    // stage A chunk: 32 rows x 64 k, coalesced float4 (x1 flat = dw row-major)
    #pragma unroll
    for (int r = 0; r < 2; ++r) {
      int idx = r * 256 + t;                 // 0..511  (512 float4)
      int m   = idx >> 4;
      int c4  = idx & 15;
      v4f v = *(const v4f*)(x1 + (size_t)m * 4096 + (size_t)kc * 64 + c4 * 4);
      *(v4f*)&Abuf[m * LDA + c4 * 4] = v;
    }
    // stage B^T chunk: 64 n-rows x 64 k, coalesced + non-temporal (604MB one-shot stream)
    #pragma unroll
    for (int r = 0; r < 4; ++r) {
      int idx = r * 256 + t;                 // 0..1023 (1024 float4)
      int nr  = idx >> 4;
      int c4  = idx & 15;
      const v4f* src = (const v4f*)(kgw + (size_t)g * OUTF * INF_ +
                                    (size_t)(n0 + nr) * INF_ + i0);
      v4f v = __builtin_nontemporal_load(src + c4);
      *(v4f*)&Bbuf[nr * LDA + c4 * 4] = v;
    }
    __syncthreads();

    // 16 wmma steps of K=4 over this chunk
    #pragma unroll
    for (int ks = 0; ks < 16; ++ks) {
      const int k0 = ks * 4 + half * 2;
      v2f a = *(const v2f*)&Abuf[arow * LDA + k0];
      v2f b = *(const v2f*)&Bbuf[brow * LDA + k0];
      acc = __builtin_amdgcn_wmma_f32_16x16x4_f32(
          /*neg_a=*/false, a, /*neg_b=*/false, b,
          /*c_mod=*/(short)0, acc, /*reuse_a=*/false, /*reuse_b=*/false);
    }
    __syncthreads();
  }

  // epilogue: +bias, store. C tile map: VGPR j -> M = j + 8*half, N = l15.
  const int n = n0 + nt * 16 + l15;
  const float bias = kgb[n] + kgb[OUTF + n] + kgb[2 * OUTF + n] + kgb[3 * OUTF + n];
  #pragma unroll
  for (int j = 0; j < 8; ++j) {
    const int m = mt * 16 + j + half * 8;    // batch index
    ksum[(size_t)m * OUTF + n] = acc[j] + bias;
  }
}

// ---------------------------------------------------------------------------
// Kernel 2: per-sample dynamic 3x3 conv + sigmoid channel gate ->
//           fused[b,co,p] = out1 + x2 * sigmoid(conv1x1(x1))
// grid = (8 co-groups, 32 batches), 256 threads = pixels
// ---------------------------------------------------------------------------
__global__ __launch_bounds__(256) void dynconv_gate(
    const float* __restrict__ x1, const float* __restrict__ x2,
    const float* __restrict__ cww, const float* __restrict__ cwb,
    const float* __restrict__ ksum, float* __restrict__ fused)
{
  __shared__ float x2s[C2 * PIX];        // 64 KB
  __shared__ float x1s[16 * PIX];        // 16 KB
  __shared__ float ks [8 * 576];         // 18 KB: 8 co's worth of 64x3x3 kernels

  const int t   = threadIdx.x;
  const int b   = blockIdx.y;
  const int co0 = blockIdx.x * 8;

  {
    const v4f* src = (const v4f*)(x2 + (size_t)b * C2 * PIX);
    v4f* dst = (v4f*)x2s;
    #pragma unroll
    for (int r = 0; r < 16; ++r) dst[r * 256 + t] = src[r * 256 + t];
  }
  {
    const v4f* src = (const v4f*)(x1 + (size_t)b * 16 * PIX);
    v4f* dst = (v4f*)x1s;
    #pragma unroll
    for (int r = 0; r < 4; ++r) dst[r * 256 + t] = src[r * 256 + t];
  }
  for (int i = t; i < 8 * 576; i += 256)
    ks[i] = ksum[(size_t)b * OUTF + (size_t)co0 * 576 + i];
  __syncthreads();

  const int p = t;
  const int y = p >> 4;
  const int x = p & 15;

  for (int cc = 0; cc < 8; ++cc) {
    const int co = co0 + cc;
    float out1 = 0.f;
    for (int ci = 0; ci < C2; ++ci) {
      const float* wk = &ks[cc * 576 + ci * 9];
      const float* xp = &x2s[ci * PIX + p];
      #pragma unroll
      for (int dy = -1; dy <= 1; ++dy) {
        #pragma unroll
        for (int dx = -1; dx <= 1; ++dx) {
          const bool ok = ((unsigned)(y + dy) < 16u) && ((unsigned)(x + dx) < 16u);
          const float v = ok ? xp[dy * 16 + dx] : 0.f;
          out1 += wk[(dy + 1) * 3 + (dx + 1)] * v;
        }
      }
    }
    float cwv = cwb[co];
    #pragma unroll
    for (int c = 0; c < 16; ++c) cwv += x1s[c * PIX + p] * cww[co * 16 + c];
    const float sig = 1.f / (1.f + __expf(-cwv));
    fused[((size_t)b * C2 + co) * PIX + p] = out1 + x2s[co * PIX + p] * sig;
  }
}

// ---------------------------------------------------------------------------
// Kernel 3: o1 = conv1x1(fused), o2/o3 = conv1x1(x1); concatenated output
// grid = 32 batches, 256 threads = pixels
// ---------------------------------------------------------------------------
#define OFF2 (BB * C2 * PIX)               // 524288
#define OFF3 (OFF2 + BB * C1 * PIX)        // 557056

__global__ __launch_bounds__(256) void outconvs(
    const float* __restrict__ x1, const float* __restrict__ fused,
    const float* __restrict__ o1w, const float* __restrict__ o1b,
    const float* __restrict__ o2w, const float* __restrict__ o2b,
    const float* __restrict__ o3w, const float* __restrict__ o3b,
    float* __restrict__ out)
{
  __shared__ float fs [C2 * PIX];          // 64 KB
  __shared__ float x1s[16 * PIX];          // 16 KB

  const int t = threadIdx.x;
  const int b = blockIdx.x;
  {
    const v4f* src = (const v4f*)(fused + (size_t)b * C2 * PIX);
    v4f* dst = (v4f*)fs;
    #pragma unroll
    for (int r = 0; r < 16; ++r) dst[r * 256 + t] = src[r * 256 + t];
  }
  {
    const v4f* src = (const v4f*)(x1 + (size_t)b * 16 * PIX);
    v4f* dst = (v4f*)x1s;
    #pragma unroll
    for (int r = 0; r < 4; ++r) dst[r * 256 + t] = src[r * 256 + t];
  }
  __syncthreads();

  const int p = t;
  for (int o = 0; o < C2; ++o) {
    float acc = o1b[o];
    #pragma unroll 8
    for (int co = 0; co < C2; ++co) acc += fs[co * PIX + p] * o1w[o * C2 + co];
    out[((size_t)b * C2 + o) * PIX + p] = acc;
  }
  #pragma unroll
  for (int o = 0; o < C1; ++o) {
    float acc = o2b[o];
    #pragma unroll
    for (int c = 0; c < 16; ++c) acc += x1s[c * PIX + p] * o2w[o * 16 + c];
    out[OFF2 + ((size_t)b * C1 + o) * PIX + p] = acc;
  }
  #pragma unroll
  for (int o = 0; o < 3 * C1; ++o) {
    float acc = o3b[o];
    #pragma unroll
    for (int c = 0; c < 16; ++c) acc += x1s[c * PIX + p] * o3w[o * 16 + c];
    out[OFF3 + ((size_t)b * 3 * C1 + o) * PIX + p] = acc;
  }
}

// ---------------------------------------------------------------------------
extern "C" void kernel_launch(void* const* d_in, const int* in_sizes, int n_in,
                              void* d_out, int out_size, void* d_ws, size_t ws_size,
                              hipStream_t stream) {
  (void)in_sizes; (void)n_in; (void)out_size; (void)ws_size;

  const float* x1  = (const float*)d_in[0];
  const float* x2  = (const float*)d_in[1];
  const float* kgw = (const float*)d_in[2];
  const float* kgb = (const float*)d_in[3];
  const float* cww = (const float*)d_in[4];
  const float* cwb = (const float*)d_in[5];
  const float* o1w = (const float*)d_in[6];
  const float* o1b = (const float*)d_in[7];
  const float* o2w = (const float*)d_in[8];
  const float* o2b = (const float*)d_in[9];
  const float* o3w = (const float*)d_in[10];
  const float* o3b = (const float*)d_in[11];
  float* out = (float*)d_out;

  float* ksum  = (float*)d_ws;                       // 32*36864 floats (4.72 MB)
  float* fused = ksum + (size_t)BB * OUTF;           // 32*64*256 floats (2 MB)

  ksum_gemm_wmma<<<dim3(OUTF / 64), 256, 0, stream>>>(x1, kgw, kgb, ksum);
  dynconv_gate <<<dim3(8, BB),      256, 0, stream>>>(x1, x2, cww, cwb, ksum, fused);
  outconvs     <<<dim3(BB),         256, 0, stream>>>(x1, fused, o1w, o1b, o2w, o2b,
                                                      o3w, o3b, out);
}